// DetectionLoss_14654428414652
// MI455X (gfx1250) — compile-verified
//
#include <hip/hip_runtime.h>
#include <math.h>

#define NUM_CLASSES 91
#define B_  16
#define H_  80
#define W_  80
#define A_  3
#define C_  96            // 5 + NUM_CLASSES
#define N_  64
#define BHW (B_*H_*W_)    // 102400
#define BN  (B_*N_)       // 1024
#define NOOBJ_BLOCKS (BHW/256)   // 400

typedef float v2f __attribute__((ext_vector_type(2)));
typedef float v8f __attribute__((ext_vector_type(8)));

__device__ __forceinline__ float sigmoidf_(float x) {
    return 1.0f / (1.0f + expf(-x));
}
// jax.nn.softplus(x) = max(x,0) + log1p(exp(-|x|))
__device__ __forceinline__ float softplusf_(float x) {
    return fmaxf(x, 0.0f) + log1pf(expf(-fabsf(x)));
}
// -max(log1p(-sigmoid(z)*m), -100)
__device__ __forceinline__ float noobj_term(float z, float m) {
    float p = sigmoidf_(z);
    return -fmaxf(log1pf(-p * m), -100.0f);
}

__device__ __forceinline__ void box_cell(const float* bx, int& gx, int& gy,
                                         float& tx, float& ty, float& tw, float& th) {
    float cx = (bx[0] + bx[2]) * 0.5f;
    float cy = (bx[1] + bx[3]) * 0.5f;
    float w  = bx[2] - bx[0];
    float h  = bx[3] - bx[1];
    gx = (int)floorf(cx * (float)W_); if (gx > W_ - 1) gx = W_ - 1;
    gy = (int)floorf(cy * (float)H_); if (gy > H_ - 1) gy = H_ - 1;
    tx = cx * (float)W_ - (float)gx;
    ty = cy * (float)H_ - (float)gy;
    tw = w * (float)W_;
    th = h * (float)H_;
}

// ---- kernel 1: scatter obj0 mask (duplicate writes of 1 are benign) ----
__global__ void scatter_mask_kernel(const float* __restrict__ boxes,
                                    unsigned char* __restrict__ mask) {
    int t = blockIdx.x * blockDim.x + threadIdx.x;
    if (t >= BN) return;
    int b = t / N_;
    int gx, gy; float tx, ty, tw, th;
    box_cell(boxes + (size_t)t * 4, gx, gy, tx, ty, tw, th);
    mask[(b * H_ + gy) * W_ + gx] = 1;
}

// ---- kernel 2: noobj confidence map + deterministic block reduce ----
__global__ __launch_bounds__(256)
void noobj_kernel(const float* __restrict__ pred,
                  const unsigned char* __restrict__ mask,
                  float* __restrict__ partials) {
    __shared__ float sdata[256];
    int idx = blockIdx.x * 256 + threadIdx.x;          // grid covers BHW exactly
    size_t base = (size_t)idx * (A_ * C_);
    float m0 = 1.0f - (float)mask[idx];
    float z0 = pred[base + 4];
    float z1 = pred[base + 4 + C_];
    float z2 = pred[base + 4 + 2 * C_];
    float t = noobj_term(z0, m0) + noobj_term(z1, 1.0f) + noobj_term(z2, 1.0f);
    sdata[threadIdx.x] = t;
    __syncthreads();
    for (int s = 128; s > 0; s >>= 1) {
        if (threadIdx.x < s) sdata[threadIdx.x] += sdata[threadIdx.x + s];
        __syncthreads();
    }
    if (threadIdx.x == 0) partials[blockIdx.x] = sdata[0];
}

// ---- kernel 3: positive-box terms (per-thread map, reduced later) ----
__global__ __launch_bounds__(256)
void pos_kernel(const float* __restrict__ pred,
                const float* __restrict__ boxes,
                const int* __restrict__ labels,
                float* __restrict__ posTerms) {   // layout [4][BN]
    int t = blockIdx.x * blockDim.x + threadIdx.x;
    if (t >= BN) return;
    int b = t / N_;
    int gx, gy; float tx, ty, tw, th;
    box_cell(boxes + (size_t)t * 4, gx, gy, tx, ty, tw, th);
    const float* cell = pred + (size_t)((b * H_ + gy) * W_ + gx) * (A_ * C_); // anchor 0

    float px = sigmoidf_(cell[0]);
    float py = sigmoidf_(cell[1]);
    float pw = cell[2];
    float ph = cell[3];
    float lxy = (px - tx) * (px - tx) + (py - ty) * (py - ty);
    float lwh = (pw - tw) * (pw - tw) + (ph - th) * (ph - th);

    float pconf = sigmoidf_(cell[4]);
    float cpos = -fmaxf(logf(pconf), -100.0f);

    int lab = labels[t];
    float posw = (lab == 1) ? 10.0f : 1.0f;
    float lcls = 0.0f;
    for (int k = 0; k < NUM_CLASSES; k++) {
        float z = cell[5 + k];
        lcls += (k == lab) ? posw * softplusf_(-z) : softplusf_(z);
    }
    posTerms[0 * BN + t] = lxy;
    posTerms[1 * BN + t] = lwh;
    posTerms[2 * BN + t] = cpos;
    posTerms[3 * BN + t] = lcls;
}

// ---- kernel 4: exact f32 WMMA reduction (V_WMMA_F32_16X16X4_F32) + finalize ----
// A = ones(16x4)  =>  D[m][n] = sum_k B[k][n] + C[m][n]  (pure f32 column sum).
// 4 chained WMMAs fold 256 partials -> 16 column sums (replicated in all D rows),
// then 16 shuffles finish the sum. Deterministic and bit-exact in f32.
__global__ __launch_bounds__(256)
void reduce_final_kernel(const float* __restrict__ noobjPart,
                         const float* __restrict__ posTerms,
                         float* __restrict__ out) {
    __shared__ float sdata[256];
    __shared__ float totals[5];
    int tid = threadIdx.x;

    const float* srcs[5] = { noobjPart,
                             posTerms + 0 * BN, posTerms + 1 * BN,
                             posTerms + 2 * BN, posTerms + 3 * BN };
    const int counts[5] = { NOOBJ_BLOCKS, BN, BN, BN, BN };

    for (int c = 0; c < 5; c++) {
        float v = 0.0f;
        const float* p = srcs[c];
        int n = counts[c];
        for (int i = tid; i < n; i += 256) v += p[i];
        sdata[tid] = v;
        __syncthreads();

        if (tid < 32) {                      // wave 0, EXEC all ones
            int lane = tid;
            v8f acc = {0.f, 0.f, 0.f, 0.f, 0.f, 0.f, 0.f, 0.f};
            v2f a;  a.x = 1.0f; a.y = 1.0f;  // A = ones(16x4)
#pragma unroll
            for (int j = 0; j < 4; j++) {
                v2f bm;
                bm.x = sdata[j * 64 + lane * 2 + 0];
                bm.y = sdata[j * 64 + lane * 2 + 1];
                acc = __builtin_amdgcn_wmma_f32_16x16x4_f32(
                        /*neg_a=*/false, a, /*neg_b=*/false, bm,
                        /*c_mod=*/(short)0, acc,
                        /*reuse_a=*/false, /*reuse_b=*/false);
            }
            float col = acc[0];              // lane l holds colsum(l % 16)
            float tot = 0.0f;
#pragma unroll
            for (int s = 0; s < 16; s++) tot += __shfl(col, s, 32);
            if (lane == 0) totals[c] = tot;
        }
        __syncthreads();
    }

    if (tid == 0) {
        float Sno = totals[0], Sxy = totals[1], Swh = totals[2];
        float Scp = totals[3], Scls = totals[4];
        const float inv_pos = 1.0f / (float)BN;
        float loss_xy  = Sxy * inv_pos;
        float loss_wh  = Swh * inv_pos;
        float loss_cls = Scls * inv_pos;
        float loss_conf = (Scp + 0.5f * Sno) * (1.0f / (float)(B_ * H_ * W_ * A_));
        out[0] = 5.0f * loss_xy + 5.0f * loss_wh + loss_conf + loss_cls;
        out[1] = loss_xy;
        out[2] = loss_wh;
        out[3] = loss_conf;
        out[4] = loss_cls;
    }
}

extern "C" void kernel_launch(void* const* d_in, const int* in_sizes, int n_in,
                              void* d_out, int out_size, void* d_ws, size_t ws_size,
                              hipStream_t stream) {
    const float* pred   = (const float*)d_in[0];
    const float* boxes  = (const float*)d_in[1];
    const int*   labels = (const int*)d_in[2];
    float* out = (float*)d_out;

    // workspace layout: [mask: BHW bytes][noobj partials: 400 f32][pos terms: 4*BN f32]
    unsigned char* mask   = (unsigned char*)d_ws;
    float* noobjPart      = (float*)((char*)d_ws + BHW);       // offset 102400 (4-aligned)
    float* posTerms       = noobjPart + NOOBJ_BLOCKS;

    hipMemsetAsync(mask, 0, BHW, stream);
    scatter_mask_kernel<<<(BN + 255) / 256, 256, 0, stream>>>(boxes, mask);
    noobj_kernel<<<NOOBJ_BLOCKS, 256, 0, stream>>>(pred, mask, noobjPart);
    pos_kernel<<<(BN + 255) / 256, 256, 0, stream>>>(pred, boxes, labels, posTerms);
    reduce_final_kernel<<<1, 256, 0, stream>>>(noobjPart, posTerms, out);
}